// GCN_45973329936641
// MI455X (gfx1250) — compile-verified
//
#include <hip/hip_runtime.h>

// ---------------------------------------------------------------------------
// GCN forward for MI455X (gfx1250):
//   - f32 WMMA (v_wmma_f32_16x16x4_f32) for all dense GEMMs
//   - Tensor Data Mover (tensor_load_to_lds + s_wait_tensorcnt) stages each
//     weight matrix into LDS once per workgroup; B fragments come from ds_load
//   - float4 coalesced gather + global_atomic_add_f32 for the edge scatter
// ---------------------------------------------------------------------------

typedef __attribute__((ext_vector_type(2))) float        v2f;
typedef __attribute__((ext_vector_type(8))) float        v8f;
typedef __attribute__((ext_vector_type(4))) unsigned int u32x4;
typedef __attribute__((ext_vector_type(8))) int          i32x8;
typedef __attribute__((ext_vector_type(4))) int          i32x4;

#define N_NODES  50000
#define N_EDGES  600000
#define N_GRAPHS 2500
#define F_IN     11
#define NG_PAD   2512   // 157 * 16  (graph count padded to a multiple of 16)

// ---------------- TDM: DMA a dense [rows x width] f32 block into LDS --------
// D# group0: count=1, lds_addr, 57-bit global addr, type=2.
// D# group1: data_size=2 (4B), tensor_dim0=width, tensor_dim1=rows,
//            tile_dim0=width, tile_dim1=rows, tensor_dim0_stride=width.
// Issued by one wave; completion via TENSORcnt, published with a barrier.
__device__ __forceinline__ void tdm_load_2d_to_lds(void* lds_dst, const void* gsrc,
                                                   int width, int rows) {
  unsigned           lds_off = (unsigned)(uintptr_t)lds_dst;   // addr[31:0] = LDS byte offset
  unsigned long long ga      = (unsigned long long)(uintptr_t)gsrc;
  u32x4 g0;
  g0.x = 1u;                                                   // count=1, user descriptor
  g0.y = lds_off;
  g0.z = (unsigned)ga;                                         // global_addr[31:0]
  g0.w = (unsigned)((ga >> 32) & 0x01FFFFFFull) | (2u << 30);  // global_addr[56:32] | type=2
  i32x8 g1;
  g1[0] = 0x00020000;        // workgroup_mask=0 | data_size=2 (4 bytes), no pad/iterate
  g1[1] = width << 16;       // atomic_barrier_addr=0 | tensor_dim0[15:0]
  g1[2] = rows  << 16;       // tensor_dim0[31:16]=0 | tensor_dim1[15:0]
  g1[3] = width << 16;       // tensor_dim1[31:16]=0 | tile_dim0
  g1[4] = rows;              // tile_dim1 | tile_dim2=0
  g1[5] = width;             // tensor_dim0_stride[31:0]
  g1[6] = 0;                 // stride[47:32]=0 | tensor_dim1_stride[15:0]=0
  g1[7] = 0;
  i32x4 g2 = {0, 0, 0, 0}, g3 = {0, 0, 0, 0};
#if __clang_major__ >= 23
  i32x8 g4 = {0, 0, 0, 0, 0, 0, 0, 0};
  __builtin_amdgcn_tensor_load_to_lds(g0, g1, g2, g3, g4, 0);
#else
  __builtin_amdgcn_tensor_load_to_lds(g0, g1, g2, g3, 0);
#endif
  __builtin_amdgcn_s_wait_tensorcnt(0);
}

// ---------------- degree / gcn_norm ----------------------------------------
__global__ void k_init_deg(float* __restrict__ dis) {
  int n = blockIdx.x * blockDim.x + threadIdx.x;
  if (n < N_NODES) dis[n] = 1.0f;          // self-loop contributes 1
}

__global__ void k_accum_deg(const int* __restrict__ dst, float* __restrict__ dis) {
  int e = blockIdx.x * blockDim.x + threadIdx.x;
  if (e < N_EDGES) atomicAdd(&dis[dst[e]], 1.0f);
}

__global__ void k_finalize_dis(float* __restrict__ dis) {
  int n = blockIdx.x * blockDim.x + threadIdx.x;
  if (n < N_NODES) dis[n] = rsqrtf(dis[n]);   // deg >= 1 always (self-loop)
}

// ---------------- f32 WMMA GEMM with TDM-staged B in LDS --------------------
// C[Mtiles*16, N] = A[M,K] @ B[K,N] (+bias, relu). ldb == ldc == N (dense).
// One wave computes one 16x16 tile; whole B lives in LDS (K*N*4 <= 32KB).
template <int K, int N, bool BIAS, bool RELU>
__global__ void k_gemm_wmma(const float* __restrict__ A, const float* __restrict__ B,
                            const float* __restrict__ bias, float* __restrict__ C,
                            int Mtiles, int lda) {
  constexpr int Ntiles = N / 16;
  __shared__ float sB[K * N];

  int wave = threadIdx.x >> 5;
  if (wave == 0) tdm_load_2d_to_lds(sB, B, N, K);   // one TDM op per block
  __syncthreads();

  int tile = blockIdx.x * (blockDim.x >> 5) + wave;
  if (tile >= Mtiles * Ntiles) return;              // wave-uniform exit
  int tm = tile / Ntiles;
  int tn = tile % Ntiles;

  int lane = threadIdx.x & 31;
  int half = lane >> 4;      // 0: K=k0,k0+1   1: K=k0+2,k0+3
  int l    = lane & 15;
  int arow = tm * 16 + l;    // A: lane (mod 16) = row
  int bcol = tn * 16 + l;    // B: lane (mod 16) = col

  v8f c = {0.f, 0.f, 0.f, 0.f, 0.f, 0.f, 0.f, 0.f};

#pragma unroll
  for (int k0 = 0; k0 < K; k0 += 4) {
    int kb = k0 + half * 2;
    v2f a, b;
    if constexpr (K % 4 == 0) {
      a.x = A[arow * lda + kb];
      a.y = A[arow * lda + kb + 1];
      b.x = sB[kb * N + bcol];
      b.y = sB[(kb + 1) * N + bcol];
    } else {
      // clamp addresses + mask values: no EXEC-divergent predicated loads
      bool in0 = (kb < K), in1 = (kb + 1 < K);
      int  i0  = in0 ? kb : 0, i1 = in1 ? kb + 1 : 0;
      float a0 = A[arow * lda + i0], a1 = A[arow * lda + i1];
      float b0 = sB[i0 * N + bcol],  b1 = sB[i1 * N + bcol];
      a.x = in0 ? a0 : 0.0f;  a.y = in1 ? a1 : 0.0f;
      b.x = in0 ? b0 : 0.0f;  b.y = in1 ? b1 : 0.0f;
    }
    c = __builtin_amdgcn_wmma_f32_16x16x4_f32(false, a, false, b,
                                              (short)0, c, false, false);
  }

#pragma unroll
  for (int i = 0; i < 8; ++i) {            // C/D: VGPR i -> rows i and i+8
    int crow = tm * 16 + i + half * 8;
    float v = c[i];
    if (BIAS) v += bias[bcol];
    if (RELU) v = fmaxf(v, 0.0f);
    C[crow * N + bcol] = v;
  }
}

// ---------------- message passing (float4 gather + scatter-add) -------------
template <int F>
__global__ void k_self_msgs4(const float* __restrict__ h, const float* __restrict__ dis,
                             float* __restrict__ out) {
  constexpr int FQ = F / 4;
  int t = blockIdx.x * blockDim.x + threadIdx.x;
  if (t >= N_NODES * FQ) return;
  int n = t / FQ;
  float w = dis[n] * dis[n];
  float4 v = ((const float4*)h)[t];
  ((float4*)out)[t] = make_float4(w * v.x, w * v.y, w * v.z, w * v.w);
}

template <int F>
__global__ void k_edge_msgs4(const int* __restrict__ src, const int* __restrict__ dst,
                             const float* __restrict__ dis, const float* __restrict__ h,
                             float* __restrict__ out) {
  constexpr int FQ = F / 4;
  int t = blockIdx.x * blockDim.x + threadIdx.x;
  if (t >= N_EDGES * FQ) return;
  int e  = t / FQ;
  int f4 = (t % FQ) * 4;
  int s = src[e];
  int d = dst[e];
  float  w  = dis[s] * dis[d];
  float4 hv = *(const float4*)(h + s * F + f4);   // b128 gather, wave-coalesced
  float* o  = out + d * F + f4;
  atomicAdd(o + 0, w * hv.x);
  atomicAdd(o + 1, w * hv.y);
  atomicAdd(o + 2, w * hv.z);
  atomicAdd(o + 3, w * hv.w);
}

template <int F>
__global__ void k_bias_relu4(float* __restrict__ out, const float* __restrict__ bias) {
  constexpr int FQ = F / 4;
  int t = blockIdx.x * blockDim.x + threadIdx.x;
  if (t >= N_NODES * FQ) return;
  int f4 = (t % FQ) * 4;
  float4 v = ((const float4*)out)[t];
  v.x = fmaxf(v.x + bias[f4 + 0], 0.0f);
  v.y = fmaxf(v.y + bias[f4 + 1], 0.0f);
  v.z = fmaxf(v.z + bias[f4 + 2], 0.0f);
  v.w = fmaxf(v.w + bias[f4 + 3], 0.0f);
  ((float4*)out)[t] = v;
}

// ---------------- global mean pool ------------------------------------------
__global__ void k_zero(float* __restrict__ p, int n) {
  int t = blockIdx.x * blockDim.x + threadIdx.x;
  if (t < n) p[t] = 0.0f;
}

__global__ void k_pool_accum4(const int* __restrict__ batch, const float* __restrict__ h,
                              float* __restrict__ gsum, float* __restrict__ gcnt) {
  int t = blockIdx.x * blockDim.x + threadIdx.x;
  if (t >= N_NODES * 16) return;          // 16 float4 chunks per 64-wide row
  int n  = t >> 4;
  int f4 = (t & 15) * 4;
  int g  = batch[n];
  float4 v = ((const float4*)h)[t];
  float* o = gsum + g * 64 + f4;
  atomicAdd(o + 0, v.x);
  atomicAdd(o + 1, v.y);
  atomicAdd(o + 2, v.z);
  atomicAdd(o + 3, v.w);
  if ((t & 15) == 0) atomicAdd(&gcnt[g], 1.0f);
}

__global__ void k_pool_div(float* __restrict__ gsum, const float* __restrict__ gcnt) {
  int t = blockIdx.x * blockDim.x + threadIdx.x;
  if (t >= NG_PAD * 64) return;
  gsum[t] /= fmaxf(gcnt[t >> 6], 1.0f);   // padded rows: 0 / 1 = 0
}

// ---------------- final linear (64 -> 1) ------------------------------------
__global__ void k_head3(const float* __restrict__ g2, const float* __restrict__ Wf3,
                        const float* __restrict__ bf3, float* __restrict__ out) {
  int g = blockIdx.x * blockDim.x + threadIdx.x;
  if (g >= N_GRAPHS) return;
  float acc = bf3[0];
#pragma unroll
  for (int f = 0; f < 64; ++f) acc += g2[g * 64 + f] * Wf3[f];
  out[g] = acc;
}

// ---------------------------------------------------------------------------
extern "C" void kernel_launch(void* const* d_in, const int* in_sizes, int n_in,
                              void* d_out, int out_size, void* d_ws, size_t ws_size,
                              hipStream_t stream) {
  const float* x     = (const float*)d_in[0];
  const int*   ei    = (const int*)  d_in[1];   // [2, E] row-major
  const int*   batch = (const int*)  d_in[2];
  const float* W1    = (const float*)d_in[3];
  const float* b1    = (const float*)d_in[4];
  const float* W2    = (const float*)d_in[5];
  const float* b2    = (const float*)d_in[6];
  const float* Wf1   = (const float*)d_in[7];
  const float* bf1   = (const float*)d_in[8];
  const float* Wf2   = (const float*)d_in[9];
  const float* bf2   = (const float*)d_in[10];
  const float* Wf3   = (const float*)d_in[11];
  const float* bf3   = (const float*)d_in[12];

  const int* src = ei;             // edge_index[0]
  const int* dst = ei + N_EDGES;   // edge_index[1]

  // ---- workspace layout (floats) ----
  float* ws   = (float*)d_ws;
  float* bufA = ws;                         // 50000*128   (h1, later h2)
  float* bufB = bufA + N_NODES * 128;       // 50000*128   (out1, later out2)
  float* dis  = bufB + N_NODES * 128;       // 50000       (deg -> deg^-1/2)
  float* gsum = dis + N_NODES;              // NG_PAD*64   (pooled sums / means)
  float* gcnt = gsum + NG_PAD * 64;         // NG_PAD      (graph node counts)
  float* g1   = gcnt + NG_PAD;              // NG_PAD*128
  float* g2   = g1 + NG_PAD * 128;          // NG_PAD*64

  const int T = 256;
  auto cdiv = [](int a, int b) { return (a + b - 1) / b; };

  // 1) gcn_norm: deg (with self-loops) -> deg^-1/2
  k_init_deg    <<<cdiv(N_NODES, T), T, 0, stream>>>(dis);
  k_accum_deg   <<<cdiv(N_EDGES, T), T, 0, stream>>>(dst, dis);
  k_finalize_dis<<<cdiv(N_NODES, T), T, 0, stream>>>(dis);

  // 2) layer 1 transform: h1 = x @ W1   [50000,11]x[11,128]  (K padded to 12)
  k_gemm_wmma<F_IN, 128, false, false>
      <<<cdiv(3125 * 8, 8), T, 0, stream>>>(x, W1, nullptr, bufA, 3125, F_IN);

  // 3) layer 1 aggregate: out1 = scatter(norm * h1[src]) + b1, relu
  k_self_msgs4<128><<<cdiv(N_NODES * 32, T), T, 0, stream>>>(bufA, dis, bufB);
  k_edge_msgs4<128><<<cdiv(N_EDGES * 32, T), T, 0, stream>>>(src, dst, dis, bufA, bufB);
  k_bias_relu4<128><<<cdiv(N_NODES * 32, T), T, 0, stream>>>(bufB, b1);

  // 4) layer 2 transform: h2 = out1 @ W2   [50000,128]x[128,64]
  k_gemm_wmma<128, 64, false, false>
      <<<cdiv(3125 * 4, 8), T, 0, stream>>>(bufB, W2, nullptr, bufA, 3125, 128);

  // 5) layer 2 aggregate (out1 dead -> bufB reused as out2)
  k_self_msgs4<64><<<cdiv(N_NODES * 16, T), T, 0, stream>>>(bufA, dis, bufB);
  k_edge_msgs4<64><<<cdiv(N_EDGES * 16, T), T, 0, stream>>>(src, dst, dis, bufA, bufB);
  k_bias_relu4<64><<<cdiv(N_NODES * 16, T), T, 0, stream>>>(bufB, b2);

  // 6) global mean pool into tile-padded [NG_PAD, 64]
  k_zero       <<<cdiv(NG_PAD * 65, T), T, 0, stream>>>(gsum, NG_PAD * 65); // gsum+gcnt
  k_pool_accum4<<<cdiv(N_NODES * 16, T), T, 0, stream>>>(batch, bufB, gsum, gcnt);
  k_pool_div   <<<cdiv(NG_PAD * 64, T), T, 0, stream>>>(gsum, gcnt);

  // 7) MLP head: two WMMA GEMMs with fused bias+relu, then 64->1 dot
  k_gemm_wmma<64, 128, true, true>
      <<<cdiv(157 * 8, 8), T, 0, stream>>>(gsum, Wf1, bf1, g1, 157, 64);
  k_gemm_wmma<128, 64, true, true>
      <<<cdiv(157 * 4, 8), T, 0, stream>>>(g1, Wf2, bf2, g2, 157, 128);
  k_head3<<<cdiv(N_GRAPHS, T), T, 0, stream>>>(g2, Wf3, bf3, (float*)d_out);
}